// AGGEdgeGraph_26766236188677
// MI455X (gfx1250) — compile-verified
//
#include <hip/hip_runtime.h>
#include <hip/hip_bf16.h>

#define NE 100000
#define IN_FEATS 256
#define HIDDEN 128
#define NK 16
#define ROW_TILES (NE / 16)        // 6250 16-row tiles
#define WAVES_PER_BLOCK 8

typedef float v2f __attribute__((ext_vector_type(2)));
typedef float v8f __attribute__((ext_vector_type(8)));

// ---------------------------------------------------------------------------
// Kernel 1: t = edge_feats @ W + b  via V_WMMA_F32_16X16X4_F32
//   Each wave computes a 16x128 strip (8 N-tiles of 16x16), K = 256 in steps
//   of 4. W is staged transposed into LDS so a B fragment is one ds_load_b64.
// ---------------------------------------------------------------------------
__global__ __launch_bounds__(256) void gemm_bias_wmma(
    const float* __restrict__ ef,     // [NE, IN_FEATS]
    const float* __restrict__ W,      // [IN_FEATS, HIDDEN]
    const float* __restrict__ bias,   // [HIDDEN]
    float* __restrict__ t)            // [NE, HIDDEN]
{
    // W transposed: sWt[n*IN_FEATS + k] = W[k*HIDDEN + n]   (128 KB LDS)
    __shared__ float sWt[IN_FEATS * HIDDEN];

    const int tid = threadIdx.x;
    for (int i = tid; i < IN_FEATS * HIDDEN; i += 256) {
        int k = i >> 7;        // row of W (k index)
        int n = i & (HIDDEN - 1);
        sWt[n * IN_FEATS + k] = W[i];   // coalesced global read
    }
    __syncthreads();

    const int wave   = tid >> 5;
    const int lane   = tid & 31;
    const int tile   = blockIdx.x * WAVES_PER_BLOCK + wave;
    if (tile >= ROW_TILES) return;      // wave-uniform: EXEC all-1s for WMMA

    const int row0   = tile * 16;
    const int lane15 = lane & 15;
    const int khalf  = (lane >> 4) * 2;   // lanes 0-15 -> K+{0,1}, 16-31 -> K+{2,3}

    // A fragment base: lane holds row (row0 + lane15), two consecutive K vals
    const float* arow = ef + (size_t)(row0 + lane15) * IN_FEATS + khalf;

    v8f acc[8];
#pragma unroll
    for (int nt = 0; nt < 8; ++nt) acc[nt] = (v8f){0.f,0.f,0.f,0.f,0.f,0.f,0.f,0.f};

    for (int k = 0; k < IN_FEATS; k += 4) {
        v2f a = *(const v2f*)(arow + k);              // A 16x4 fragment
#pragma unroll
        for (int nt = 0; nt < 8; ++nt) {
            const int col = nt * 16 + lane15;
            // B 4x16 fragment: lane holds col, two consecutive K vals
            v2f bf = *(const v2f*)(&sWt[col * IN_FEATS + k + khalf]);
            acc[nt] = __builtin_amdgcn_wmma_f32_16x16x4_f32(
                false, a, false, bf, (short)0, acc[nt], false, false);
        }
    }

    // Epilogue: C/D layout -> lane L, VGPR j holds (row0 + j + 8*(L>=16), col)
    const int hi = lane >> 4;
#pragma unroll
    for (int nt = 0; nt < 8; ++nt) {
        const int col = nt * 16 + lane15;
        const float bv = bias[col];
#pragma unroll
        for (int j = 0; j < 8; ++j) {
            const int row = row0 + j + hi * 8;
            t[(size_t)row * HIDDEN + col] = acc[nt][j] + bv;
        }
    }
}

// ---------------------------------------------------------------------------
// Kernel 2: out[e] = t[e] + sum_k t[nbr[e][k]]
//   One wave (32 lanes) per edge; each lane owns one float4 of the 128-wide
//   hidden vector. t (51 MB) lives in L2 (192 MB) so gathers stay on-chip.
// ---------------------------------------------------------------------------
__global__ __launch_bounds__(256) void aggregate_edges(
    const float* __restrict__ t,      // [NE, HIDDEN]
    const int*   __restrict__ nbr,    // [NE, NK]
    float*       __restrict__ out)    // [NE, HIDDEN]
{
    const int gid = blockIdx.x * 256 + threadIdx.x;
    const int e   = gid >> 5;          // one wave per edge
    const int seg = gid & 31;          // float4 segment within hidden dim
    if (e >= NE) return;

    const float4* t4   = (const float4*)t;
    float4*       out4 = (float4*)out;

    float4 acc = t4[(size_t)e * (HIDDEN / 4) + seg];

    const int* ni = nbr + (size_t)e * NK;
#pragma unroll
    for (int k = 0; k < NK; ++k) {
        const int idx = ni[k];                         // same cacheline per wave
        const float4 v = t4[(size_t)idx * (HIDDEN / 4) + seg];  // coalesced b128
        acc.x += v.x; acc.y += v.y; acc.z += v.z; acc.w += v.w;
    }
    out4[(size_t)e * (HIDDEN / 4) + seg] = acc;
}

// ---------------------------------------------------------------------------
extern "C" void kernel_launch(void* const* d_in, const int* in_sizes, int n_in,
                              void* d_out, int out_size, void* d_ws, size_t ws_size,
                              hipStream_t stream) {
    const float* edge_feats = (const float*)d_in[0];   // [NE, IN_FEATS]
    const int*   neighbors  = (const int*)d_in[1];     // [NE, NK]
    const float* W          = (const float*)d_in[2];   // [IN_FEATS, HIDDEN]
    const float* b          = (const float*)d_in[3];   // [HIDDEN]
    float*       out        = (float*)d_out;           // [NE, HIDDEN]
    float*       t          = (float*)d_ws;            // scratch [NE, HIDDEN] = 51.2 MB

    // GEMM: 6250 row-tiles, 8 waves/block
    const int gemm_blocks = (ROW_TILES + WAVES_PER_BLOCK - 1) / WAVES_PER_BLOCK;
    gemm_bias_wmma<<<gemm_blocks, 256, 0, stream>>>(edge_feats, W, b, t);

    // Aggregation: NE waves of 32 lanes
    const int agg_threads = NE * 32;
    const int agg_blocks  = (agg_threads + 255) / 256;
    aggregate_edges<<<agg_blocks, 256, 0, stream>>>(t, neighbors, out);
}